// Group_46273977647728
// MI455X (gfx1250) — compile-verified
//
#include <hip/hip_runtime.h>
#include <hip/hip_bf16.h>

// ---------------------------------------------------------------------------
// Problem constants (from reference): B=8, N=8192, NUM_GROUP=512, GROUP_SIZE=32
// Outputs (flat, in return order):
//   neighborhood: (B, 512, 32, 3) f32   -> 393216 floats
//   center      : (B, 512, 3)    f32   ->  12288 floats
// ---------------------------------------------------------------------------
#define BATCH 8
#define NPTS 8192
#define NGRP 512
#define GRPSZ 32

typedef float v2f __attribute__((ext_vector_type(2)));
typedef float v8f __attribute__((ext_vector_type(8)));

#define BIGF 3.0e38f

// 32-lane argmax (max value; tie -> lower lane's index wins) -- all lanes
// converge to the same (v, l).
__device__ __forceinline__ void argmax32(float& v, int& l) {
#pragma unroll
  for (int off = 16; off; off >>= 1) {
    float ov = __shfl_xor(v, off);
    int ol = __shfl_xor(l, off);
    if (ov > v || (ov == v && ol < l)) { v = ov; l = ol; }
  }
}

// ---------------------------------------------------------------------------
// Kernel 1: SE(3) transform; also precompute ||x||^2 -> float4 (x,y,z,n2)
// ---------------------------------------------------------------------------
__global__ __launch_bounds__(256) void se3_kernel(const float* __restrict__ xyz,
                                                  const float* __restrict__ pose,
                                                  float4* __restrict__ x4) {
  int id = blockIdx.x * blockDim.x + threadIdx.x;
  if (id >= BATCH * NPTS) return;
  int b = id >> 13;  // / NPTS
  const float* p = xyz + (size_t)id * 3;
  const float* P = pose + (size_t)b * 12;
  float X = p[0], Y = p[1], Z = p[2];
  float x = P[0] * X + P[1] * Y + P[2]  * Z + P[3];
  float y = P[4] * X + P[5] * Y + P[6]  * Z + P[7];
  float z = P[8] * X + P[9] * Y + P[10] * Z + P[11];
  x4[id] = make_float4(x, y, z, x * x + y * y + z * z);
}

// ---------------------------------------------------------------------------
// Kernel 2: farthest point sampling. One persistent block per batch.
// 1024 threads (32 waves), 8 points/thread kept entirely in registers.
// Matches jax scan semantics: out[0] = 0; out[g] = argmax after point out[g-1].
// argmax tie-break = first (lowest) index, like jnp.argmax.
// ---------------------------------------------------------------------------
__global__ __launch_bounds__(1024) void fps_kernel(const float4* __restrict__ x4,
                                                   float* __restrict__ center) {
  __shared__ float4 s_last;
  __shared__ float s_v[32];
  __shared__ int s_i[32];

  const int tid = threadIdx.x;
  const int b = blockIdx.x;
  const float4* pts = x4 + (size_t)b * NPTS;

  float xs[8], ys[8], zs[8], nn[8], dist[8];
#pragma unroll
  for (int i = 0; i < 8; ++i) {
    int p = tid + (i << 10);
    float4 q = pts[p];
    xs[i] = q.x; ys[i] = q.y; zs[i] = q.z; nn[i] = q.w;
    dist[i] = 1.0e10f;
  }

  if (tid == 0) {
    float4 q = pts[0];
    s_last = q;
    float* co = center + (size_t)(b * NGRP + 0) * 3;
    co[0] = q.x; co[1] = q.y; co[2] = q.z;
  }
  __syncthreads();

  const int lane = tid & 31;
  const int wv = tid >> 5;

  for (int g = 1; g < NGRP; ++g) {
    float4 L = s_last;
    float best = -BIGF;
    int bi = 0;
#pragma unroll
    for (int i = 0; i < 8; ++i) {
      float d = nn[i] + L.w - 2.0f * (xs[i] * L.x + ys[i] * L.y + zs[i] * L.z);
      d = fminf(dist[i], d);
      dist[i] = d;
      if (d > best) { best = d; bi = tid + (i << 10); }
    }
    argmax32(best, bi);
    if (lane == 0) { s_v[wv] = best; s_i[wv] = bi; }
    __syncthreads();
    if (wv == 0) {
      float v = s_v[lane];
      int ii = s_i[lane];
      argmax32(v, ii);
      if (lane == 0) {
        float4 q = pts[ii];
        s_last = q;
        float* co = center + (size_t)(b * NGRP + g) * 3;
        co[0] = q.x; co[1] = q.y; co[2] = q.z;
      }
    }
    __syncthreads();
  }
}

// ---------------------------------------------------------------------------
// Kernel 3: KNN via V_WMMA_F32_16X16X4_F32, 4 tiles (64 points) in flight.
// One wave per (batch, tile of 16 centers): grid = 8*32 = 256, block = 32.
//
// A (16x4, f32): row m = (-2cx, -2cy, -2cz, 1); lanes 0-15 hold K={0,1},
// lanes 16-31 hold K={2,3} (ISA 16x4 A layout). Built once, reused 512x.
// B (4x16, f32): col n = (x, y, z, ||x||^2); same half-lane K split.
// D[m][n] = -2 c.x + ||x||^2 ; + ||c||^2 per row -> true squared distance.
//
// Per-row top-32 kept distributed (1 slot/lane). The row's current worst
// (max) entry and its owner lane are CACHED in registers and recomputed
// (5-shuffle argmax) only when an insertion actually lands, so the
// steady-state per-tile cost is one compare + one ballot per row.
// Final 32-lane bitonic sort ascending (matches jax.lax.top_k order).
// ---------------------------------------------------------------------------
__global__ __launch_bounds__(32) void knn_kernel(const float4* __restrict__ x4,
                                                 const float* __restrict__ center,
                                                 float* __restrict__ nbr) {
  const int lane = threadIdx.x;
  const int b = blockIdx.x >> 5;
  const int g0 = (blockIdx.x & 31) << 4;  // 16 centers per tile
  const float4* pts = x4 + (size_t)b * NPTS;
  const float* cbase = center + (size_t)(b * NGRP + g0) * 3;

  __shared__ float s_cn[16];

  const int m = lane & 15;
  float cx = cbase[m * 3 + 0];
  float cy = cbase[m * 3 + 1];
  float cz = cbase[m * 3 + 2];
  if (lane < 16) s_cn[m] = cx * cx + cy * cy + cz * cz;
  __syncthreads();

  v2f a;
  if (lane < 16) { a[0] = -2.0f * cx; a[1] = -2.0f * cy; }
  else           { a[0] = -2.0f * cz; a[1] = 1.0f; }

  float rn[8];
  const int rbase = (lane < 16) ? 0 : 8;
#pragma unroll
  for (int v = 0; v < 8; ++v) rn[v] = s_cn[rbase + v];

  float Ld[16];  // this lane's slot of row r's 32-entry list
  int Li[16];
  float mv[16];  // cached: current max of row r's list (uniform)
  int ml[16];    // cached: lane owning that max (uniform)
#pragma unroll
  for (int r = 0; r < 16; ++r) { Ld[r] = BIGF; Li[r] = 0; mv[r] = BIGF; ml[r] = 0; }

  for (int t = 0; t < NPTS / 64; ++t) {
    const int pbase = t * 64 + (lane & 15);
    float4 P0 = pts[pbase];
    float4 P1 = pts[pbase + 16];
    float4 P2 = pts[pbase + 32];
    float4 P3 = pts[pbase + 48];
    if (t + 1 < NPTS / 64)
      __builtin_prefetch((const void*)&pts[pbase + 64], 0, 3);

    v8f cc[4];
    {
      v2f b0, b1, b2, b3;
      if (lane < 16) {
        b0[0] = P0.x; b0[1] = P0.y;
        b1[0] = P1.x; b1[1] = P1.y;
        b2[0] = P2.x; b2[1] = P2.y;
        b3[0] = P3.x; b3[1] = P3.y;
      } else {
        b0[0] = P0.z; b0[1] = P0.w;
        b1[0] = P1.z; b1[1] = P1.w;
        b2[0] = P2.z; b2[1] = P2.w;
        b3[0] = P3.z; b3[1] = P3.w;
      }
      v8f z0 = {}, z1 = {}, z2 = {}, z3 = {};
      cc[0] = __builtin_amdgcn_wmma_f32_16x16x4_f32(false, a, false, b0, (short)0, z0, false, false);
      cc[1] = __builtin_amdgcn_wmma_f32_16x16x4_f32(false, a, false, b1, (short)0, z1, false, false);
      cc[2] = __builtin_amdgcn_wmma_f32_16x16x4_f32(false, a, false, b2, (short)0, z2, false, false);
      cc[3] = __builtin_amdgcn_wmma_f32_16x16x4_f32(false, a, false, b3, (short)0, z3, false, false);
    }

#pragma unroll
    for (int u = 0; u < 4; ++u) {
      const int p = pbase + u * 16;  // this lane's column point index
#pragma unroll
      for (int r = 0; r < 16; ++r) {
        const bool mine = ((lane >= 16) == (r >= 8));
        float d = mine ? (cc[u][r & 7] + rn[r & 7]) : BIGF;
        unsigned long long ins = __ballot(mine && d < mv[r]);
        while (ins) {
          int src = (int)__ffsll(ins) - 1;  // uniform -> v_readlane
          ins &= ins - 1;
          float cd = __int_as_float(
              __builtin_amdgcn_readlane(__float_as_int(d), src));
          int ci = __builtin_amdgcn_readlane(p, src);
          if (cd < mv[r]) {
            if (lane == ml[r]) { Ld[r] = cd; Li[r] = ci; }
            float nv = Ld[r];
            int nl = lane;
            argmax32(nv, nl);
            mv[r] = nv;
            ml[r] = nl;
          }
        }
      }
    }
  }

  // Sort each row's 32 entries ascending by distance and emit gathered points.
  for (int r = 0; r < 16; ++r) {
    float d = Ld[r];
    int i = Li[r];
#pragma unroll
    for (int k = 2; k <= 32; k <<= 1) {
#pragma unroll
      for (int j = k >> 1; j > 0; j >>= 1) {
        float od = __shfl_xor(d, j);
        int oi = __shfl_xor(i, j);
        bool takeMin = (((lane & j) == 0) == ((lane & k) == 0));
        bool sw = takeMin ? (od < d || (od == d && oi < i))
                          : (od > d || (od == d && oi > i));
        if (sw) { d = od; i = oi; }
      }
    }
    float4 P = pts[i];
    float* o = nbr + (size_t)(((b * NGRP + g0 + r) * GRPSZ) + lane) * 3;
    o[0] = P.x; o[1] = P.y; o[2] = P.z;
  }
}

// ---------------------------------------------------------------------------
extern "C" void kernel_launch(void* const* d_in, const int* in_sizes, int n_in,
                              void* d_out, int out_size, void* d_ws, size_t ws_size,
                              hipStream_t stream) {
  (void)in_sizes; (void)n_in; (void)out_size; (void)ws_size;
  const float* xyz = (const float*)d_in[0];    // (B, N, 3) f32
  const float* pose = (const float*)d_in[1];   // (B, 3, 4) f32
  float* out = (float*)d_out;
  float* nbr = out;                                        // (B,512,32,3)
  float* center = out + (size_t)BATCH * NGRP * GRPSZ * 3;  // (B,512,3)
  float4* x4 = (float4*)d_ws;                              // (B,N) float4 scratch

  se3_kernel<<<(BATCH * NPTS + 255) / 256, 256, 0, stream>>>(xyz, pose, x4);
  fps_kernel<<<BATCH, 1024, 0, stream>>>(x4, center);
  knn_kernel<<<BATCH * (NGRP / 16), 32, 0, stream>>>(x4, center, nbr);
}